// TunableQuantizer_57320633532915
// MI455X (gfx1250) — compile-verified
//
#include <hip/hip_runtime.h>
#include <stdint.h>

#define HIDDEN   4096
#define POD      64
#define ROWLEN   (HIDDEN + POD)   /* 4160 */
#define QLEV     7.0f

typedef unsigned int u32;
typedef __attribute__((ext_vector_type(4))) unsigned int v4u;
typedef __attribute__((ext_vector_type(8))) int          v8i;
typedef __attribute__((ext_vector_type(4))) int          v4i;

__device__ __forceinline__ float fmax3(float a, float b, float c) {
    return fmaxf(fmaxf(a, b), c);     // -> v_max3_num_f32 (abs folded as modifiers)
}

// quantize one float: round-to-nearest-even, clamp to [-128,127], return i32 (low byte = int8)
__device__ __forceinline__ u32 q1(float x, float inv) {
    float v = __builtin_amdgcn_fmed3f(rintf(x * inv), -128.0f, 127.0f);
    return (u32)(int)v;               // v_cvt_i32_f32 (exact: v is integral)
}

// pack 4 quantized dwords' low bytes into one dword with v_perm_b32
__device__ __forceinline__ u32 pack4(float4 f, float inv) {
    u32 a = q1(f.x, inv), b = q1(f.y, inv), c = q1(f.z, inv), d = q1(f.w, inv);
    u32 lo = __builtin_amdgcn_perm(b, a, 0x0c0c0400u);  // [a0, b0, 0, 0]
    u32 hi = __builtin_amdgcn_perm(d, c, 0x0c0c0400u);  // [c0, d0, 0, 0]
    return  __builtin_amdgcn_perm(hi, lo, 0x05040100u); // [a0, b0, c0, d0]
}

__global__ __launch_bounds__(256)
void quantize_pod_kernel(const float* __restrict__ x,
                         int8_t*   __restrict__ qout,
                         _Float16* __restrict__ scales,
                         float*    __restrict__ outliers)
{
    __shared__ float tile[HIDDEN];   // 16 KB row tile (TDM destination)
    __shared__ float wred[8];        // per-wave max-abs

    const int row = blockIdx.x;
    const int t   = threadIdx.x;
    const float* src = x + (size_t)row * ROWLEN;

    // ---- Wave 0: DMA the 4096-float quantize segment into LDS via TDM ----
    if (t < 32) {
        uint64_t ga  = (uint64_t)(uintptr_t)(src + POD);
        u32      lds = (u32)(uintptr_t)(void*)tile;   // LDS byte offset

        v4u g0;
        g0[0] = 1u;                                   // count=1, user-mode, no gather
        g0[1] = lds;                                  // lds_addr
        g0[2] = (u32)(ga & 0xFFFFFFFFu);              // global_addr[31:0]
        g0[3] = (u32)((ga >> 32) & 0x01FFFFFFu)       // global_addr[56:32]
              | (2u << 30);                           // type=2 ("image")

        v8i g1;
        g1[0] = (2 << 16);            // data_size=2 (4 bytes); mask=0; no pad/iterate
        g1[1] = (int)(HIDDEN << 16);  // tensor_dim0[15:0]=4096 in bits[63:48]
        g1[2] = (1 << 16);            // tensor_dim0 hi=0; tensor_dim1[15:0]=1
        g1[3] = (int)(HIDDEN << 16);  // tile_dim0=4096 in bits[127:112]
        g1[4] = 1;                    // tile_dim1=1, tile_dim2=0
        g1[5] = HIDDEN;               // tensor_dim0_stride low32 = 4096
        g1[6] = 0;
        g1[7] = 0;

        v4i g2 = (v4i){0, 0, 0, 0};   // <=2D tensor: groups 2/3 unused
        v4i g3 = (v4i){0, 0, 0, 0};
        v8i g4 = (v8i){0, 0, 0, 0, 0, 0, 0, 0};  // extra group (clang-23 6-arg form)

        __builtin_amdgcn_tensor_load_to_lds(g0, g1, g2, g3, g4, 0);
    }

    // ---- Overlap: copy the 64 outlier floats while the DMA is in flight ----
    if (t < 16) {
        const float4* osrc = (const float4*)src;                   // 16B-aligned (row*16640)
        float4*       odst = (float4*)(outliers + (size_t)row * POD);
        odst[t] = osrc[t];
    }

    __builtin_amdgcn_s_wait_tensorcnt(0);
    __syncthreads();

    // ---- Read 16 floats/thread from LDS, local max|.| via max3 chains ----
    const float4* tf = (const float4*)(tile + t * 16);
    float4 f0 = tf[0], f1 = tf[1], f2 = tf[2], f3 = tf[3];

    float m;
    m = fmax3(fabsf(f0.x), fabsf(f0.y), fabsf(f0.z));
    m = fmax3(m, fabsf(f0.w), fabsf(f1.x));
    m = fmax3(m, fabsf(f1.y), fabsf(f1.z));
    m = fmax3(m, fabsf(f1.w), fabsf(f2.x));
    m = fmax3(m, fabsf(f2.y), fabsf(f2.z));
    m = fmax3(m, fabsf(f2.w), fabsf(f3.x));
    m = fmax3(m, fabsf(f3.y), fabsf(f3.z));
    m = fmaxf(m, fabsf(f3.w));

    // ---- wave32 butterfly reduce, then cross-wave via LDS ----
    #pragma unroll
    for (int off = 16; off > 0; off >>= 1)
        m = fmaxf(m, __shfl_xor(m, off, 32));

    if ((t & 31) == 0) wred[t >> 5] = m;
    __syncthreads();

    float gm = fmax3(fmax3(wred[0], wred[1], wred[2]),
                     fmax3(wred[3], wred[4], wred[5]),
                     fmaxf(wred[6], wred[7]));

    // ---- scale: fp16-round max/7 (INPUT_CLIP_RATIO == 1.0) ----
    _Float16 sh = (_Float16)(gm / QLEV);
    float    sf = (float)sh;
    float    inv = 1.0f / sf;
    if (t == 0) scales[row] = sh;

    // ---- quantize 16 values, pack to 16 bytes, one non-temporal B128 store ----
    v4u packed;
    packed[0] = pack4(f0, inv);
    packed[1] = pack4(f1, inv);
    packed[2] = pack4(f2, inv);
    packed[3] = pack4(f3, inv);
    __builtin_nontemporal_store(packed, (v4u*)(qout + (size_t)row * HIDDEN + t * 16));
}

extern "C" void kernel_launch(void* const* d_in, const int* in_sizes, int n_in,
                              void* d_out, int out_size, void* d_ws, size_t ws_size,
                              hipStream_t stream)
{
    (void)n_in; (void)out_size; (void)d_ws; (void)ws_size;

    const float* x = (const float*)d_in[0];
    const int rows = in_sizes[0] / ROWLEN;   // 16384 for the reference shapes

    // d_out = q(int8) [rows*4096] ++ scales(fp16) [rows] ++ outliers(fp32) [rows*64]
    int8_t*   q        = (int8_t*)d_out;
    size_t    qbytes   = (size_t)rows * HIDDEN;
    _Float16* scales   = (_Float16*)((char*)d_out + qbytes);
    float*    outliers = (float*)((char*)d_out + qbytes + (size_t)rows * sizeof(_Float16));

    quantize_pod_kernel<<<rows, 256, 0, stream>>>(x, q, scales, outliers);
}